// GIBGCN_13134009991725
// MI455X (gfx1250) — compile-verified
//
#include <hip/hip_runtime.h>
#include <math.h>

// ---------------------------------------------------------------------------
// Problem constants (from reference)
// ---------------------------------------------------------------------------
#define N_NODES   50000
#define N_EDGES   800000
#define F_IN      512
#define LATENT    64
#define N_CLASSES 16
#define ROW_TILES (N_NODES / 16)   // 3125, exact

typedef __attribute__((ext_vector_type(2))) float v2f;
typedef __attribute__((ext_vector_type(8))) float v8f;

// ---------------------------------------------------------------------------
// CDNA5 async global->LDS copy (ASYNCcnt-tracked, no VGPR staging)
// ---------------------------------------------------------------------------
__device__ __forceinline__ void async_b128_to_lds(unsigned lds_off, const void* gptr) {
    asm volatile("global_load_async_to_lds_b128 %0, %1, off"
                 :: "v"(lds_off), "v"(gptr)
                 : "memory");
}
__device__ __forceinline__ void wait_async_all() {
    asm volatile("s_wait_asynccnt 0x0" ::: "memory");
}

// ---------------------------------------------------------------------------
// Degree / norm kernels
// ---------------------------------------------------------------------------
__global__ void k_init_deg(float* __restrict__ deg) {
    int i = blockIdx.x * blockDim.x + threadIdx.x;
    if (i < N_NODES) deg[i] = 1.0f;              // self-loop
}

__global__ void k_edge_deg(const int* __restrict__ dst, float* __restrict__ deg) {
    int e = blockIdx.x * blockDim.x + threadIdx.x;
    if (e < N_EDGES) atomicAdd(&deg[dst[e]], 1.0f);
}

__global__ void k_dinv(const float* __restrict__ deg, float* __restrict__ dinv) {
    int i = blockIdx.x * blockDim.x + threadIdx.x;
    if (i < N_NODES) dinv[i] = rsqrtf(deg[i]);
}

// ---------------------------------------------------------------------------
// GEMM1: h1[50000x64] = x[50000x512] @ W1[512x64]  (f32 WMMA 16x16x4)
// The 16x512 A-tile (32 KB) is staged once per block into LDS via async
// global->LDS B128 copies, then shared by all 4 waves (one 16-col slice each).
// ---------------------------------------------------------------------------
__global__ void k_gemm1_wmma(const float* __restrict__ x,
                             const float* __restrict__ W1,
                             float* __restrict__ h1) {
    __shared__ float As[16 * F_IN];              // 32 KB A-tile

    const int row0 = blockIdx.x * 16;            // 16-row tile
    const int col0 = (threadIdx.x >> 5) * 16;    // wave -> 16-col slice
    const int lane = threadIdx.x & 31;
    const int m    = lane & 15;                  // M (A) / N (B,D) index
    const int kk   = (lane >> 4) << 1;           // K pair: 0 or 2

    // ---- async stage A-tile: 2048 B128 packets, 16 per thread ----
    {
        const float* xbase = x + (size_t)row0 * F_IN;
        const unsigned lds0 = (unsigned)(uintptr_t)As;
        #pragma unroll
        for (int it = 0; it < 16; ++it) {
            int idx = threadIdx.x + it * 128;    // float4 index in tile
            async_b128_to_lds(lds0 + idx * 16, xbase + idx * 4);
        }
        wait_async_all();                        // own wave's copies done
        __syncthreads();                         // all waves' copies visible
    }

    const float* arow = As + m * F_IN + kk;                 // LDS
    const float* bcol = W1 + (size_t)kk * LATENT + col0 + m; // global (L2-hot)

    v8f c = {};
    #pragma unroll 4
    for (int k = 0; k < F_IN; k += 4) {
        v2f a = *(const v2f*)(arow + k);         // ds_load_b64
        v2f b;
        b.x = bcol[(size_t)k * LATENT];          // B[k+kk][n]
        b.y = bcol[(size_t)(k + 1) * LATENT];    // B[k+kk+1][n]
        c = __builtin_amdgcn_wmma_f32_16x16x4_f32(
                false, a, false, b, (short)0, c, false, false);
    }

    // D layout: VGPR r -> row (lane>>4)*8 + r, col lane&15
    const int mb = (lane >> 4) << 3;
    #pragma unroll
    for (int r = 0; r < 8; ++r)
        h1[(size_t)(row0 + mb + r) * LATENT + col0 + m] = c[r];
}

// ---------------------------------------------------------------------------
// GEMM2: h2[50000x16] = out1[50000x64] @ W2[64x16]  (f32 WMMA 16x16x4)
// N=16 -> one wave per row tile; 4 row tiles per 128-thread block.
// ---------------------------------------------------------------------------
__global__ void k_gemm2_wmma(const float* __restrict__ out1,
                             const float* __restrict__ W2,
                             float* __restrict__ h2) {
    const int tile = blockIdx.x * 4 + (threadIdx.x >> 5);
    if (tile >= ROW_TILES) return;               // wave-uniform guard
    const int row0 = tile * 16;
    const int lane = threadIdx.x & 31;
    const int m    = lane & 15;
    const int kk   = (lane >> 4) << 1;

    const float* arow = out1 + (size_t)(row0 + m) * LATENT + kk;
    const float* bcol = W2   + (size_t)kk * N_CLASSES + m;

    v8f c = {};
    #pragma unroll
    for (int k = 0; k < LATENT; k += 4) {
        v2f a = *(const v2f*)(arow + k);
        v2f b;
        b.x = bcol[(size_t)k * N_CLASSES];
        b.y = bcol[(size_t)(k + 1) * N_CLASSES];
        c = __builtin_amdgcn_wmma_f32_16x16x4_f32(
                false, a, false, b, (short)0, c, false, false);
    }

    const int mb = (lane >> 4) << 3;
    #pragma unroll
    for (int r = 0; r < 8; ++r)
        h2[(size_t)(row0 + mb + r) * N_CLASSES + m] = c[r];
}

// ---------------------------------------------------------------------------
// Self-loop + bias initialization:  out[i][f] = h[i][f]*dinv[i]^2 + b[f]
// ---------------------------------------------------------------------------
template <int F>
__global__ void k_self_bias(const float* __restrict__ h,
                            const float* __restrict__ dinv,
                            const float* __restrict__ b,
                            float* __restrict__ out) {
    int t = blockIdx.x * blockDim.x + threadIdx.x;
    if (t >= N_NODES * F) return;
    int i = t / F, f = t % F;
    float di = dinv[i];
    out[t] = h[t] * di * di + b[f];
}

// ---------------------------------------------------------------------------
// Edge scatter:  out[dst][f] += h[src][f] * dinv[src]*dinv[dst]
// One thread handles 4 contiguous features (float4 gather, 4 f32 atomics
// resolved in the 192 MB L2 — h and out both fit easily).
// ---------------------------------------------------------------------------
template <int F>
__global__ void k_scatter(const int* __restrict__ src,
                          const int* __restrict__ dst,
                          const float* __restrict__ h,
                          const float* __restrict__ dinv,
                          float* __restrict__ out) {
    constexpr int GROUPS = F / 4;
    long long t = (long long)blockIdx.x * blockDim.x + threadIdx.x;
    if (t >= (long long)N_EDGES * GROUPS) return;
    int e  = (int)(t / GROUPS);
    int f4 = (int)(t % GROUPS) * 4;
    int s = src[e], d = dst[e];
    float w = dinv[s] * dinv[d];
    const float4 v = *(const float4*)(h + (size_t)s * F + f4);
    float* o = out + (size_t)d * F + f4;
    atomicAdd(o + 0, v.x * w);
    atomicAdd(o + 1, v.y * w);
    atomicAdd(o + 2, v.z * w);
    atomicAdd(o + 3, v.w * w);
}

// ---------------------------------------------------------------------------
// KL head:  mean = out[:, :F/2], std = softplus(out[:, F/2:]) + 1e-10
//           kl   = -log(std) + 0.5*(std^2 + mean^2) - 0.5 ; skl = 0
// ---------------------------------------------------------------------------
__device__ __forceinline__ float softplus_f(float v) {
    return (v > 20.0f) ? v : log1pf(__expf(v));
}

template <int F>
__global__ void k_kl(const float* __restrict__ out,
                     float* __restrict__ ixz,
                     float* __restrict__ skl) {
    constexpr int H = F / 2;
    int t = blockIdx.x * blockDim.x + threadIdx.x;
    if (t >= N_NODES * H) return;
    int i = t / H, j = t % H;
    float mean = out[(size_t)i * F + j];
    float std  = softplus_f(out[(size_t)i * F + H + j]) + 1e-10f;
    ixz[t] = -logf(std) + 0.5f * (std * std + mean * mean) - 0.5f;
    skl[t] = 0.0f;
}

// ---------------------------------------------------------------------------
// Launcher
// ---------------------------------------------------------------------------
extern "C" void kernel_launch(void* const* d_in, const int* in_sizes, int n_in,
                              void* d_out, int out_size, void* d_ws, size_t ws_size,
                              hipStream_t stream) {
    const float* x  = (const float*)d_in[0];
    const int*   ei = (const int*)d_in[1];      // (2, N_EDGES): row0=src, row1=dst
    const float* W1 = (const float*)d_in[2];
    const float* b1 = (const float*)d_in[3];
    const float* W2 = (const float*)d_in[4];
    const float* b2 = (const float*)d_in[5];
    const int* src = ei;
    const int* dst = ei + N_EDGES;

    // Workspace layout (floats)
    float* ws   = (float*)d_ws;
    float* deg  = ws;                                   // 50000
    float* dinv = ws + N_NODES;                         // 50000
    float* h1   = ws + 2 * N_NODES;                     // 50000*64
    float* h2   = h1 + (size_t)N_NODES * LATENT;        // 50000*16

    // Output layout (floats, concatenated in reference return order)
    float* out  = (float*)d_out;
    float* out2 = out;                                  // 50000*16
    float* out1 = out2 + (size_t)N_NODES * N_CLASSES;   // 50000*64
    float* ixz1 = out1 + (size_t)N_NODES * LATENT;      // 50000*32
    float* skl1 = ixz1 + (size_t)N_NODES * (LATENT / 2);
    float* ixz2 = skl1 + (size_t)N_NODES * (LATENT / 2);// 50000*8
    float* skl2 = ixz2 + (size_t)N_NODES * (N_CLASSES / 2);

    const int B = 256;
    // --- degrees & symmetric norm ---
    k_init_deg<<<(N_NODES + B - 1) / B, B, 0, stream>>>(deg);
    k_edge_deg<<<(N_EDGES + B - 1) / B, B, 0, stream>>>(dst, deg);
    k_dinv<<<(N_NODES + B - 1) / B, B, 0, stream>>>(deg, dinv);

    // --- layer 1 ---
    k_gemm1_wmma<<<ROW_TILES, 128, 0, stream>>>(x, W1, h1);
    {
        long long n = (long long)N_NODES * LATENT;
        k_self_bias<LATENT><<<(unsigned)((n + B - 1) / B), B, 0, stream>>>(h1, dinv, b1, out1);
        long long m = (long long)N_EDGES * (LATENT / 4);
        k_scatter<LATENT><<<(unsigned)((m + B - 1) / B), B, 0, stream>>>(src, dst, h1, dinv, out1);
        long long p = (long long)N_NODES * (LATENT / 2);
        k_kl<LATENT><<<(unsigned)((p + B - 1) / B), B, 0, stream>>>(out1, ixz1, skl1);
    }

    // --- layer 2 ---
    k_gemm2_wmma<<<(ROW_TILES + 3) / 4, 128, 0, stream>>>(out1, W2, h2);
    {
        long long n = (long long)N_NODES * N_CLASSES;
        k_self_bias<N_CLASSES><<<(unsigned)((n + B - 1) / B), B, 0, stream>>>(h2, dinv, b2, out2);
        long long m = (long long)N_EDGES * (N_CLASSES / 4);
        k_scatter<N_CLASSES><<<(unsigned)((m + B - 1) / B), B, 0, stream>>>(src, dst, h2, dinv, out2);
        long long p = (long long)N_NODES * (N_CLASSES / 2);
        k_kl<N_CLASSES><<<(unsigned)((p + B - 1) / B), B, 0, stream>>>(out2, ixz2, skl2);
    }
}